// GATPredictor_31507880084101
// MI455X (gfx1250) — compile-verified
//
#include <hip/hip_runtime.h>
#include <math.h>

// GAT predictor for MI455X (gfx1250), fp32 end-to-end.
// Dense GEMMs: V_WMMA_F32_16X16X4_F32, LDS-staged weight panels
//   (async GLOBAL_LOAD_ASYNC_TO_LDS_B128 staging when the builtin exists).
// Sparse softmax/aggregation: float atomics (L2-resident: x/agg ~102MB < 192MB L2).

#define HEADS 4
#define CHN   128
#define HCN   512
#define KB    16          // K-chunk staged in LDS per iteration
#define NEG_SLOPE 0.2f

#if defined(__AMDGCN__) && \
    __has_builtin(__builtin_amdgcn_global_load_async_to_lds_b128) && \
    __has_builtin(__builtin_amdgcn_s_wait_asynccnt)
#define USE_ASYNC_LDS 1
#else
#define USE_ASYNC_LDS 0
#endif

typedef __attribute__((ext_vector_type(2))) float v2f;
typedef __attribute__((ext_vector_type(8))) float v8f;

#if USE_ASYNC_LDS
typedef int v4i __attribute__((ext_vector_type(4)));
typedef __attribute__((address_space(1))) v4i* g1_v4i;   // global (AS1) int4*
typedef __attribute__((address_space(3))) v4i* l3_v4i;   // LDS    (AS3) int4*

// Issue one 16B async LDS-DMA transfer (ASYNCcnt-tracked).
// Global flat address == AS1 address numerically; generic pointer to a
// __shared__ object truncates to the 32-bit LDS offset (ISA 10.2), which is
// exactly the AS3 pointer representation — so integer casts are sound.
__device__ __forceinline__ void async_ld_b128(const float* g, const float* l) {
  __builtin_amdgcn_global_load_async_to_lds_b128(
      (g1_v4i)(unsigned long long)g,
      (l3_v4i)(unsigned int)(unsigned long long)l,
      0, 0);
}
#endif

__device__ __forceinline__ float leakyr(float v) { return v > 0.f ? v : NEG_SLOPE * v; }
__device__ __forceinline__ float elu_f (float v) { return v > 0.f ? v : (expf(v) - 1.f); }
__device__ __forceinline__ float gelu_f(float v) {
  float u = 0.7978845608028654f * (v + 0.044715f * v * v * v);
  return 0.5f * v * (1.f + tanhf(u));
}

// ---------------------------------------------------------------------------
// WMMA f32 16x16x4 GEMM, LDS-staged.
//   Block: 256 threads (8 waves) -> one 16-row strip x all Ncols columns.
//   Wave w owns NTPW consecutive 16-col tiles (8*NTPW*16 == Ncols).
//   Per K-chunk (KB=16): stage W[k0:k0+KB, :] (contiguous rows) and
//   A[row0:row0+16, k0:k0+KB] into LDS — via async LDS-DMA if available —
//   then 4 WMMA k-steps per chunk fed from ds_load.
// Lane mapping per CDNA5 ISA 7.12.2:
//   A frag: row=l16, float2 holds K = half*2+{0,1}
//   B frag: col=l16, K = half*2+{0,1}
//   C frag: 8 VGPRs, row = row0 + half*8 + v, col = l16
// Optional second operand pair (A2,W2,K2) implements the [h, h_res] concat.
// bias==nullptr means zero bias; act==1 applies tanh-GELU.
// ---------------------------------------------------------------------------
template<int NTPW>
__global__ void gemm_wmma_kernel(const float* __restrict__ A,  int lda,  int K,
                                 const float* __restrict__ A2, int lda2, int K2,
                                 const float* __restrict__ W,
                                 const float* __restrict__ W2,
                                 const float* __restrict__ bias,
                                 float* __restrict__ Cmat, int Ncols, int act)
{
  __shared__ float ldsW[KB * HCN];   // up to 16x512 floats = 32KB
  __shared__ float ldsA[16 * KB];    // 1KB

  const int tid  = threadIdx.x;      // 0..255, 8 waves, no divergence -> EXEC full
  const int wave = tid >> 5;
  const int lane = tid & 31;
  const int l16  = lane & 15;
  const int half = lane >> 4;
  const int row0 = blockIdx.x * 16;

  v8f acc[NTPW];
#pragma unroll
  for (int t = 0; t < NTPW; ++t) acc[t] = (v8f){};

  for (int pass = 0; pass < 2; ++pass) {
    const int Kp = pass ? K2 : K;
    if (Kp == 0) break;                       // uniform branch
    const float* Ap  = pass ? A2 : A;
    const float* Wp  = pass ? W2 : W;
    const int    ldp = pass ? lda2 : lda;

    for (int k0 = 0; k0 < Kp; k0 += KB) {
      __syncthreads();                        // protect LDS reuse across chunks
#if USE_ASYNC_LDS
      // --- async LDS-DMA staging (ASYNCcnt-tracked, no VGPR round trip) ---
      for (int i = tid * 4; i < KB * Ncols; i += 256 * 4)
        async_ld_b128(&Wp[(size_t)k0 * Ncols + i], &ldsW[i]);
      for (int i = tid; i < 16 * (KB / 4); i += 256) {
        const int r = i / (KB / 4);
        const int o = (i % (KB / 4)) * 4;
        async_ld_b128(&Ap[(size_t)(row0 + r) * ldp + k0 + o], &ldsA[r * KB + o]);
      }
      __builtin_amdgcn_s_wait_asynccnt(0);
#else
      // --- synchronous staging fallback ---
      for (int i = tid * 4; i < KB * Ncols; i += 256 * 4)
        *(float4*)&ldsW[i] = *(const float4*)&Wp[(size_t)k0 * Ncols + i];
      for (int i = tid; i < 16 * (KB / 4); i += 256) {
        const int r = i / (KB / 4);
        const int o = (i % (KB / 4)) * 4;
        *(float4*)&ldsA[r * KB + o] =
            *(const float4*)&Ap[(size_t)(row0 + r) * ldp + k0 + o];
      }
#endif
      // --- prefetch next W chunk toward the WGP cache ---
      if (k0 + KB < Kp && tid * 4 < KB * Ncols)
        __builtin_prefetch(&Wp[(size_t)(k0 + KB) * Ncols + tid * 4], 0, 3);
      __syncthreads();

#pragma unroll
      for (int kk = 0; kk < KB; kk += 4) {
        const int ka = kk + half * 2;
        v2f a = *(const v2f*)&ldsA[l16 * KB + ka];
#pragma unroll
        for (int t = 0; t < NTPW; ++t) {
          const int col = (wave * NTPW + t) * 16 + l16;
          v2f b;
          b.x = ldsW[ka * Ncols + col];
          b.y = ldsW[(ka + 1) * Ncols + col];
          acc[t] = __builtin_amdgcn_wmma_f32_16x16x4_f32(
              false, a, false, b, (short)0, acc[t], false, false);
        }
      }
    }
  }

  const int rbase = row0 + half * 8;
#pragma unroll
  for (int t = 0; t < NTPW; ++t) {
    const int col = (wave * NTPW + t) * 16 + l16;
    const float bc = bias ? bias[col] : 0.f;
#pragma unroll
    for (int v = 0; v < 8; ++v) {
      float val = acc[t][v] + bc;
      if (act == 1) val = gelu_f(val);
      Cmat[(size_t)(rbase + v) * Ncols + col] = val;
    }
  }
}

// ---------------------------------------------------------------------------
// alpha_src/alpha_dst per (node, head) + m seeded with the self-loop logit.
// ---------------------------------------------------------------------------
__global__ void alpha_kernel(const float* __restrict__ x,
                             const float* __restrict__ a_src,
                             const float* __restrict__ a_dst,
                             float* __restrict__ asrc, float* __restrict__ adst,
                             float* __restrict__ m, int Nn)
{
  const int wid  = (int)((blockIdx.x * (size_t)blockDim.x + threadIdx.x) >> 5);
  const int lane = threadIdx.x & 31;
  if (wid >= Nn * HEADS) return;             // wave-uniform guard
  const int n = wid >> 2, h = wid & 3;
  const float* xr = x + (size_t)n * HCN + h * CHN;
  const float* as = a_src + h * CHN;
  const float* ad = a_dst + h * CHN;
  float sa = 0.f, sd = 0.f;
#pragma unroll
  for (int i = lane; i < CHN; i += 32) {
    const float xv = xr[i];
    sa += xv * as[i];
    sd += xv * ad[i];
  }
#pragma unroll
  for (int off = 16; off; off >>= 1) {
    sa += __shfl_down(sa, off);
    sd += __shfl_down(sd, off);
  }
  if (lane == 0) {
    asrc[wid] = sa;
    adst[wid] = sd;
    m[wid]    = leakyr(sa + sd);             // self-loop logit seeds segment max
  }
}

// Segment-max over real edges via global fmax atomics.
__global__ void edge_max_kernel(const int* __restrict__ ei,
                                const float* __restrict__ asrc,
                                const float* __restrict__ adst,
                                float* __restrict__ m, int E)
{
  const int t = blockIdx.x * blockDim.x + threadIdx.x;
  if (t >= E * HEADS) return;
  const int e = t >> 2, h = t & 3;
  const int s = ei[e], d = ei[E + e];
  const float v = leakyr(asrc[s * HEADS + h] + adst[d * HEADS + h]);
  __hip_atomic_fetch_max(&m[d * HEADS + h], v, __ATOMIC_RELAXED,
                         __HIP_MEMORY_SCOPE_AGENT);
}

// s := exp(e_self - m)
__global__ void sinit_kernel(const float* __restrict__ asrc,
                             const float* __restrict__ adst,
                             const float* __restrict__ m,
                             float* __restrict__ s, int Nn)
{
  const int t = blockIdx.x * blockDim.x + threadIdx.x;
  if (t >= Nn * HEADS) return;
  s[t] = expf(leakyr(asrc[t] + adst[t]) - m[t]);
}

// s[dst] += exp(e - m[dst]) over real edges.
__global__ void edge_sum_kernel(const int* __restrict__ ei,
                                const float* __restrict__ asrc,
                                const float* __restrict__ adst,
                                const float* __restrict__ m,
                                float* __restrict__ s, int E)
{
  const int t = blockIdx.x * blockDim.x + threadIdx.x;
  if (t >= E * HEADS) return;
  const int e = t >> 2, h = t & 3;
  const int sidx = ei[e], d = ei[E + e];
  const int nh = d * HEADS + h;
  const float v = leakyr(asrc[sidx * HEADS + h] + adst[nh]);
  atomicAdd(&s[nh], expf(v - m[nh]));
}

// agg := x * alpha_self   (full overwrite -> no zero pass)
__global__ void agg_init_kernel(const float* __restrict__ x,
                                const float* __restrict__ asrc,
                                const float* __restrict__ adst,
                                const float* __restrict__ m,
                                const float* __restrict__ s,
                                float* __restrict__ agg, int Nn)
{
  const size_t t = blockIdx.x * (size_t)blockDim.x + threadIdx.x;
  if (t >= (size_t)Nn * HCN) return;
  const int n = (int)(t >> 9), c = (int)(t & 511), h = c >> 7;
  const int nh = n * HEADS + h;
  const float es = leakyr(asrc[nh] + adst[nh]);
  const float al = expf(es - m[nh]) / (s[nh] + 1e-16f);  // self-loop ew = 1
  agg[t] = x[t] * al;
}

// agg[dst] += x[src] * alpha * ew  — 128 threads/edge, float4 per thread.
__global__ void edge_msg_kernel(const int* __restrict__ ei,
                                const float* __restrict__ ew,
                                const float* __restrict__ x,
                                const float* __restrict__ asrc,
                                const float* __restrict__ adst,
                                const float* __restrict__ m,
                                const float* __restrict__ s,
                                float* __restrict__ agg, int E)
{
  const size_t t = blockIdx.x * (size_t)blockDim.x + threadIdx.x;
  if (t >= (size_t)E * 128) return;
  const int e  = (int)(t >> 7);
  const int c  = (int)(t & 127) * 4;       // 0..508, 16B aligned
  const int h  = c >> 7;
  const int sn = ei[e], d = ei[E + e];
  const int nh = d * HEADS + h;
  const float v  = leakyr(asrc[sn * HEADS + h] + adst[nh]);
  const float al = expf(v - m[nh]) / (s[nh] + 1e-16f) * ew[e];
  const float4 xv = *(const float4*)(x + (size_t)sn * HCN + c);
  float* dst = agg + (size_t)d * HCN + c;
  atomicAdd(dst + 0, xv.x * al);
  atomicAdd(dst + 1, xv.y * al);
  atomicAdd(dst + 2, xv.z * al);
  atomicAdd(dst + 3, xv.w * al);
}

// h_next := ELU(agg + bias)
__global__ void finalize_kernel(const float* __restrict__ agg,
                                const float* __restrict__ bias,
                                float* __restrict__ hout, int Nn)
{
  const size_t t = blockIdx.x * (size_t)blockDim.x + threadIdx.x;
  if (t >= (size_t)Nn * HCN) return;
  hout[t] = elu_f(agg[t] + bias[t & 511]);
}

// out[n] = dot(z[n, 0:128], Wp2) + bp2  — one wave per node, float4 per lane.
__global__ void mv_kernel(const float* __restrict__ z,
                          const float* __restrict__ Wp2,
                          const float* __restrict__ bp2,
                          float* __restrict__ out, int Nn)
{
  const int wid  = (int)((blockIdx.x * (size_t)blockDim.x + threadIdx.x) >> 5);
  const int lane = threadIdx.x & 31;
  if (wid >= Nn) return;
  const float4 zv = *(const float4*)(z + (size_t)wid * CHN + lane * 4);
  const float4 wv = *(const float4*)(Wp2 + lane * 4);
  float acc = zv.x * wv.x + zv.y * wv.y + zv.z * wv.z + zv.w * wv.w;
#pragma unroll
  for (int off = 16; off; off >>= 1) acc += __shfl_down(acc, off);
  if (lane == 0) out[wid] = acc + bp2[0];
}

// ---------------------------------------------------------------------------
extern "C" void kernel_launch(void* const* d_in, const int* in_sizes, int n_in,
                              void* d_out, int out_size, void* d_ws, size_t ws_size,
                              hipStream_t stream)
{
  const float* X    = (const float*)d_in[0];
  const int*   EI   = (const int*)  d_in[1];
  const float* EW   = (const float*)d_in[2];
  const float* W_in = (const float*)d_in[3];
  const float* b_in = (const float*)d_in[4];
  const float* lin0 = (const float*)d_in[5];
  const float* as0  = (const float*)d_in[6];
  const float* ad0  = (const float*)d_in[7];
  const float* bi0  = (const float*)d_in[8];
  const float* lin1 = (const float*)d_in[9];
  const float* as1  = (const float*)d_in[10];
  const float* ad1  = (const float*)d_in[11];
  const float* bi1  = (const float*)d_in[12];
  const float* Wp1  = (const float*)d_in[13];
  const float* bp1  = (const float*)d_in[14];
  const float* Wp2  = (const float*)d_in[15];
  const float* bp2  = (const float*)d_in[16];
  float* out = (float*)d_out;

  const int N = in_sizes[0] / 64;   // 50000 (divisible by 16)
  const int E = in_sizes[1] / 2;    // 800000

  // Workspace carve-up (floats).
  float* ws   = (float*)d_ws;
  const size_t N128 = (size_t)N * CHN;
  const size_t N512 = (size_t)N * HCN;
  float* h0   = ws;                 // (N,128) input projection (= residual)
  float* hA   = h0   + N128;        // (N,512) layer hidden
  float* xb   = hA   + N512;        // (N,512) lin output x (reused as z)
  float* agg  = xb   + N512;        // (N,512) aggregation buffer
  float* asrc = agg  + N512;        // (N,4)
  float* adst = asrc + (size_t)N * HEADS;
  float* mseg = adst + (size_t)N * HEADS;
  float* sseg = mseg + (size_t)N * HEADS;

  const dim3 blk256(256, 1, 1);
  const int  Mt         = N / 16;                       // 3125 exact
  const int  nhBlocks   = (N * HEADS * 32 + 255) / 256; // wave per (n,h)
  const int  ehBlocks   = (E * HEADS + 255) / 256;
  const int  nhtBlocks  = (N * HEADS + 255) / 256;
  const int  elemBlocks = (int)(((size_t)N * HCN + 255) / 256);
  const int  msgBlocks  = (int)(((size_t)E * 128 + 255) / 256);
  const int  mvBlocks   = (N * 32 + 255) / 256;

  // 1) h0 = X @ W_in + b_in          (K=64, Ncols=128, NTPW=1)
  gemm_wmma_kernel<1><<<Mt, blk256, 0, stream>>>(
      X, 64, 64, nullptr, 0, 0, W_in, nullptr, b_in, h0, CHN, 0);

  // ---- GAT layer 0: x = h0 @ lin0  (no bias inside layer) ----
  gemm_wmma_kernel<4><<<Mt, blk256, 0, stream>>>(
      h0, CHN, CHN, nullptr, 0, 0, lin0, nullptr, nullptr, xb, HCN, 0);
  alpha_kernel<<<nhBlocks, blk256, 0, stream>>>(xb, as0, ad0, asrc, adst, mseg, N);
  edge_max_kernel<<<ehBlocks, blk256, 0, stream>>>(EI, asrc, adst, mseg, E);
  sinit_kernel<<<nhtBlocks, blk256, 0, stream>>>(asrc, adst, mseg, sseg, N);
  edge_sum_kernel<<<ehBlocks, blk256, 0, stream>>>(EI, asrc, adst, mseg, sseg, E);
  agg_init_kernel<<<elemBlocks, blk256, 0, stream>>>(xb, asrc, adst, mseg, sseg, agg, N);
  edge_msg_kernel<<<msgBlocks, blk256, 0, stream>>>(EI, EW, xb, asrc, adst, mseg, sseg, agg, E);
  finalize_kernel<<<elemBlocks, blk256, 0, stream>>>(agg, bi0, hA, N);

  // ---- GAT layer 1: x = hA @ lin1 ----
  gemm_wmma_kernel<4><<<Mt, blk256, 0, stream>>>(
      hA, HCN, HCN, nullptr, 0, 0, lin1, nullptr, nullptr, xb, HCN, 0);
  alpha_kernel<<<nhBlocks, blk256, 0, stream>>>(xb, as1, ad1, asrc, adst, mseg, N);
  edge_max_kernel<<<ehBlocks, blk256, 0, stream>>>(EI, asrc, adst, mseg, E);
  sinit_kernel<<<nhtBlocks, blk256, 0, stream>>>(asrc, adst, mseg, sseg, N);
  edge_sum_kernel<<<ehBlocks, blk256, 0, stream>>>(EI, asrc, adst, mseg, sseg, E);
  agg_init_kernel<<<elemBlocks, blk256, 0, stream>>>(xb, asrc, adst, mseg, sseg, agg, N);
  edge_msg_kernel<<<msgBlocks, blk256, 0, stream>>>(EI, EW, xb, asrc, adst, mseg, sseg, agg, E);
  finalize_kernel<<<elemBlocks, blk256, 0, stream>>>(agg, bi1, hA, N);

  // ---- Head: z = gelu([hA, h0] @ Wp1 + bp1)  (K1=512 + K2=128, Ncols=128) ----
  gemm_wmma_kernel<1><<<Mt, blk256, 0, stream>>>(
      hA, HCN, HCN, h0, CHN, CHN, Wp1, Wp1 + (size_t)HCN * CHN, bp1, xb, CHN, 1);

  // ---- out = z @ Wp2 + bp2 ----
  mv_kernel<<<mvBlocks, blk256, 0, stream>>>(xb, Wp2, bp2, out, N);
}